// S4DOnlyRNN_53291954209257
// MI455X (gfx1250) — compile-verified
//
#include <hip/hip_runtime.h>
#include <hip/hip_bf16.h>
#include <math.h>

// ---------------------------------------------------------------------------
// S4D scan via chunked WMMA (gfx1250 / MI455X)
//
// y[t] = gelu( 2*Re(sum_n C_n s_{t,n}) + D*x[t] ),  s_t = dA*s_{t-1} + dB*x[t]
//
// Chunked (T=16): per chunk with start-state S:
//   Y = K @ Xc + R @ S            (Toeplitz conv kernel + state readout)
//   S <- P (.) S + W @ Xc         (complex diag * state + injection)
// All matmuls are V_WMMA_F32_16X16X4_F32 chains (full f32 precision).
//
// Sequence split into NSEG=16 segments; each wave packs TWO segments into the
// 16 WMMA columns (cols 0-7 = batch of segment 2s, cols 8-15 = segment 2s+1),
// so all 16 columns / all 32 lanes are live -> no EXEC divergence, 2x WMMA
// utilization.  Pass A computes per-segment state corrections, a tiny combine
// kernel chains boundary states in-place (corr slot s-1 is reused as the
// boundary state for segment s) and writes the final complex state, pass B
// produces outputs with software-pipelined x loads.
// ---------------------------------------------------------------------------

typedef float v2f __attribute__((ext_vector_type(2)));
typedef float v8f __attribute__((ext_vector_type(8)));

static constexpr int H    = 512;   // d_model
static constexpr int N2   = 32;    // complex modes
static constexpr int Bsz  = 8;     // batch
static constexpr int L    = 4096;  // seq len
static constexpr int T    = 16;    // chunk
static constexpr int NSEG = 16;           // segments (2 per wave)
static constexpr int NSP  = NSEG / 2;     // segment-pairs (grid.y)
static constexpr int LSEG = L / NSEG;     // 256 steps per segment
static constexpr int CPS  = LSEG / T;     // 16 chunks per segment

// workspace layout (float offsets)
static constexpr size_t OFF_K  = 0;                        // [H][16][16]
static constexpr size_t OFF_R  = OFF_K  + (size_t)H*256;   // [H][16][64]
static constexpr size_t OFF_W  = OFF_R  + (size_t)H*1024;  // [H][64][16]
static constexpr size_t OFF_P  = OFF_W  + (size_t)H*1024;  // [H][32][2]
// shared corr / boundary-state buffer: [NSEG][H][64][8]
static constexpr size_t OFF_CS = OFF_P  + (size_t)H*64;
static constexpr size_t YS_SIZE = (size_t)Bsz * L * H;     // 16777216
// total ws: 5,406,720 floats ~= 21.6 MB

static __device__ __forceinline__ v8f zero8() {
    v8f z;
#pragma unroll
    for (int i = 0; i < 8; ++i) z[i] = 0.0f;
    return z;
}
static __device__ __forceinline__ v2f mk2(float a, float b) {
    v2f r; r.x = a; r.y = b; return r;
}
static __device__ __forceinline__ v2f ld2(const float* p) {
    float2 t = *reinterpret_cast<const float2*>(p);   // 8B-aligned by layout
    return mk2(t.x, t.y);
}
// D = A(16x4) * B(4x16) + C, all f32.
static __device__ __forceinline__ v8f wmma4(v2f a, v2f b, v8f c) {
    return __builtin_amdgcn_wmma_f32_16x16x4_f32(false, a, false, b,
                                                 (short)0, c, false, false);
}
// load the 4 B-operand slices of one x chunk (branchless, all lanes valid)
static __device__ __forceinline__ void load_x(const float* __restrict__ xbase,
                                              int t0, int hw, v2f* xb) {
#pragma unroll
    for (int kk = 0; kk < 4; ++kk) {
        const float* xp = xbase + (size_t)(t0 + 4 * kk + 2 * hw) * H;
        xb[kk] = mk2(xp[0], xp[H]);
    }
}

// ---------------------------------------------------------------------------
// Kernel 1: per-h constant matrices (K, R, W, P).  512 blocks x 1 wave.
// ---------------------------------------------------------------------------
__global__ void __launch_bounds__(32)
k_setup(const float* __restrict__ log_dt, const float* __restrict__ A_real_log,
        const float* __restrict__ A_imag, const float* __restrict__ B_re,
        const float* __restrict__ B_im,   const float* __restrict__ C_re,
        const float* __restrict__ C_im,   const float* __restrict__ Dvec,
        float* __restrict__ ws)
{
    const int h = blockIdx.x;
    const int n = threadIdx.x;           // mode 0..31

    __shared__ float powre[17][32];
    __shared__ float powim[17][32];
    __shared__ float contrib[16][32];
    __shared__ float hj[16];

    const float dtv = expf(log_dt[h]);
    const float Ar  = -expf(A_real_log[h * N2 + n]);
    const float Ai  = A_imag[h * N2 + n];
    const float er  = expf(dtv * Ar);
    const float dAre = er * cosf(dtv * Ai);
    const float dAim = er * sinf(dtv * Ai);

    // dB = (Br + i Bi) * (exp(dtA)-1) / A
    const float zr = dAre - 1.0f, zi = dAim;
    const float inv = 1.0f / (Ar * Ar + Ai * Ai);
    const float qr = (zr * Ar + zi * Ai) * inv;
    const float qi = (zi * Ar - zr * Ai) * inv;
    const float Br = B_re[h * N2 + n], Bi = B_im[h * N2 + n];
    const float dBre = Br * qr - Bi * qi;
    const float dBim = Br * qi + Bi * qr;
    const float Cre = C_re[h * N2 + n], Cim = C_im[h * N2 + n];

    // powers dA^0 .. dA^16
    float cr = 1.0f, ci = 0.0f;
#pragma unroll
    for (int j = 0; j <= 16; ++j) {
        powre[j][n] = cr; powim[j][n] = ci;
        const float nr = cr * dAre - ci * dAim;
        ci = cr * dAim + ci * dAre;
        cr = nr;
    }

    // P = dA^16
    ws[OFF_P + (size_t)h * 64 + 2 * n + 0] = powre[16][n];
    ws[OFF_P + (size_t)h * 64 + 2 * n + 1] = powim[16][n];

    // conv taps h_j = 2 * sum_n Re(C * dA^j * dB)
#pragma unroll
    for (int j = 0; j < 16; ++j) {
        const float ur = powre[j][n] * dBre - powim[j][n] * dBim;
        const float ui = powre[j][n] * dBim + powim[j][n] * dBre;
        contrib[j][n] = Cre * ur - Cim * ui;
    }
    __syncthreads();
    if (n < 16) {
        float s = 0.0f;
#pragma unroll
        for (int m = 0; m < 32; ++m) s += contrib[n][m];
        hj[n] = 2.0f * s;
    }
    __syncthreads();

    // K[tau][k] = (k<tau) h_{tau-k} ; diag: h_0 + D
    const float Dh = Dvec[h];
    float* Km = ws + OFF_K + (size_t)h * 256;
    for (int idx = n; idx < 256; idx += 32) {
        const int ta = idx >> 4, k = idx & 15;
        float v = 0.0f;
        if (k <= ta) { v = hj[ta - k]; if (k == ta) v += Dh; }
        Km[idx] = v;
    }

    // R[tau][2n] = 2*Re(C*dA^{tau+1}), R[tau][2n+1] = -2*Im(...)
    float* Rm = ws + OFF_R + (size_t)h * 1024;
#pragma unroll
    for (int ta = 0; ta < 16; ++ta) {
        const float pr = powre[ta + 1][n], pi = powim[ta + 1][n];
        const float Gre = Cre * pr - Cim * pi;
        const float Gim = Cre * pi + Cim * pr;
        Rm[ta * 64 + 2 * n + 0] =  2.0f * Gre;
        Rm[ta * 64 + 2 * n + 1] = -2.0f * Gim;
    }

    // W[2n][k] = Re(dA^{15-k} dB), W[2n+1][k] = Im(...)
    float* Wm = ws + OFF_W + (size_t)h * 1024;
#pragma unroll
    for (int k = 0; k < 16; ++k) {
        const float pr = powre[15 - k][n], pi = powim[15 - k][n];
        Wm[(2 * n + 0) * 16 + k] = pr * dBre - pi * dBim;
        Wm[(2 * n + 1) * 16 + k] = pr * dBim + pi * dBre;
    }
}

// ---------------------------------------------------------------------------
// Kernel 2 (pass A): per-(h, segment-pair) state corrections via WMMA.
// grid (H, NSP) x 1 wave.  cols 0-7: segment 2*sp, cols 8-15: segment 2*sp+1.
// ---------------------------------------------------------------------------
__global__ void __launch_bounds__(32)
k_passA(const float* __restrict__ x, float* __restrict__ ws)
{
    const int h   = blockIdx.x;
    const int sp  = blockIdx.y;
    const int l   = threadIdx.x;
    const int hw  = l >> 4;       // wave half
    const int cb  = l & 15;       // column / A-matrix row M
    const int b   = cb & 7;       // batch
    const int seg = 2 * sp + (cb >> 3);

    // W as 16 A-operands (4 row tiles x 4 K-slices)
    const float* Wm = ws + OFF_W + (size_t)h * 1024;
    v2f wa[4][4];
#pragma unroll
    for (int m = 0; m < 4; ++m)
#pragma unroll
        for (int kk = 0; kk < 4; ++kk)
            wa[m][kk] = ld2(Wm + (16 * m + cb) * 16 + 4 * kk + 2 * hw);

    // P values for the complex pairs this lane owns in each D tile
    float pr[4][4], pi[4][4];
#pragma unroll
    for (int m = 0; m < 4; ++m)
#pragma unroll
        for (int j = 0; j < 4; ++j) {
            v2f p = ld2(ws + OFF_P + (size_t)h * 64 + 2 * (8 * m + 4 * hw + j));
            pr[m][j] = p.x; pi[m][j] = p.y;
        }

    v8f S[4];
#pragma unroll
    for (int m = 0; m < 4; ++m) S[m] = zero8();

    const float* xbase = x + (size_t)b * L * H + h;
    const int t00 = seg * LSEG;

    v2f xb[4];
    load_x(xbase, t00, hw, xb);                 // prime the pipeline

    for (int c = 0; c < CPS; ++c) {
        v2f xn[4];
        const int cn = (c + 1 < CPS) ? c + 1 : CPS - 1;
        load_x(xbase, t00 + cn * T, hw, xn);    // prefetch next chunk

        // S <- P (.) S
#pragma unroll
        for (int m = 0; m < 4; ++m)
#pragma unroll
            for (int j = 0; j < 4; ++j) {
                const float re = S[m][2 * j], im = S[m][2 * j + 1];
                S[m][2 * j]     = pr[m][j] * re - pi[m][j] * im;
                S[m][2 * j + 1] = pr[m][j] * im + pi[m][j] * re;
            }
        // S += W @ Xc
#pragma unroll
        for (int m = 0; m < 4; ++m)
#pragma unroll
            for (int kk = 0; kk < 4; ++kk)
                S[m] = wmma4(wa[m][kk], xb[kk], S[m]);

#pragma unroll
        for (int kk = 0; kk < 4; ++kk) xb[kk] = xn[kk];
    }

    // all 32 lanes store: corr[seg][h][row][b]
    float* cm = ws + OFF_CS + ((size_t)seg * H + h) * 512;
#pragma unroll
    for (int m = 0; m < 4; ++m)
#pragma unroll
        for (int r = 0; r < 8; ++r)
            cm[(16 * m + 8 * hw + r) * 8 + b] = S[m][r];
}

// ---------------------------------------------------------------------------
// Kernel 3: chain boundary states in-place (slot s-1 := state before seg s),
// write final complex state.  grid H x 1 wave (lane = mode).
// ---------------------------------------------------------------------------
__global__ void __launch_bounds__(32)
k_combine(const float* __restrict__ log_dt, const float* __restrict__ A_real_log,
          const float* __restrict__ A_imag, float* __restrict__ ws,
          float* __restrict__ out)
{
    const int h = blockIdx.x;
    const int n = threadIdx.x;

    const float dtv = expf(log_dt[h]);
    const float Ar  = -expf(A_real_log[h * N2 + n]);
    const float Ai  = A_imag[h * N2 + n];
    const float arg = (float)LSEG * dtv;
    const float er  = expf(arg * Ar);
    const float Pr  = er * cosf(arg * Ai);
    const float Pi  = er * sinf(arg * Ai);

    float Sre[Bsz], Sim[Bsz];
#pragma unroll
    for (int b = 0; b < Bsz; ++b) { Sre[b] = 0.0f; Sim[b] = 0.0f; }

    for (int s = 0; s < NSEG; ++s) {
        float* slot = ws + OFF_CS + ((size_t)s * H + h) * 512;
        // 1) read corr[s]
        float cr_[Bsz], ci_[Bsz];
#pragma unroll
        for (int b = 0; b < Bsz; ++b) {
            cr_[b] = slot[(2 * n + 0) * 8 + b];
            ci_[b] = slot[(2 * n + 1) * 8 + b];
        }
        // 2) publish state-before-segment-s into slot s-1 (consumed already)
        if (s > 0) {
            float* prev = ws + OFF_CS + ((size_t)(s - 1) * H + h) * 512;
#pragma unroll
            for (int b = 0; b < Bsz; ++b) {
                prev[(2 * n + 0) * 8 + b] = Sre[b];
                prev[(2 * n + 1) * 8 + b] = Sim[b];
            }
        }
        // 3) S <- Pseg (.) S + corr[s]
#pragma unroll
        for (int b = 0; b < Bsz; ++b) {
            const float nr = Pr * Sre[b] - Pi * Sim[b] + cr_[b];
            Sim[b] = Pr * Sim[b] + Pi * Sre[b] + ci_[b];
            Sre[b] = nr;
        }
    }
    // final_state (B,H,N2) complex64 appended after ys
#pragma unroll
    for (int b = 0; b < Bsz; ++b) {
        const size_t o = YS_SIZE + (((size_t)b * H + h) * N2 + n) * 2;
        out[o + 0] = Sre[b];
        out[o + 1] = Sim[b];
    }
}

// ---------------------------------------------------------------------------
// Kernel 4 (pass B): outputs.  grid (H, NSP) x 1 wave, 2 segments per wave.
// ---------------------------------------------------------------------------
__global__ void __launch_bounds__(32)
k_passB(const float* __restrict__ x, float* __restrict__ ws,
        float* __restrict__ out)
{
    __shared__ float sbuf[16][64];   // [col][row] state mirror for B-operands

    const int h   = blockIdx.x;
    const int sp  = blockIdx.y;
    const int l   = threadIdx.x;
    const int hw  = l >> 4;
    const int cb  = l & 15;
    const int b   = cb & 7;
    const int seg = 2 * sp + (cb >> 3);

    const float* Km = ws + OFF_K + (size_t)h * 256;
    const float* Rm = ws + OFF_R + (size_t)h * 1024;
    const float* Wm = ws + OFF_W + (size_t)h * 1024;

    v2f ka[4], ra[16], wa[4][4];
#pragma unroll
    for (int kk = 0; kk < 4; ++kk)
        ka[kk] = ld2(Km + cb * 16 + 4 * kk + 2 * hw);
#pragma unroll
    for (int kk = 0; kk < 16; ++kk)
        ra[kk] = ld2(Rm + cb * 64 + 4 * kk + 2 * hw);
#pragma unroll
    for (int m = 0; m < 4; ++m)
#pragma unroll
        for (int kk = 0; kk < 4; ++kk)
            wa[m][kk] = ld2(Wm + (16 * m + cb) * 16 + 4 * kk + 2 * hw);

    float pr[4][4], pi[4][4];
#pragma unroll
    for (int m = 0; m < 4; ++m)
#pragma unroll
        for (int j = 0; j < 4; ++j) {
            v2f p = ld2(ws + OFF_P + (size_t)h * 64 + 2 * (8 * m + 4 * hw + j));
            pr[m][j] = p.x; pi[m][j] = p.y;
        }

    // segment start state (slot seg-1; segment 0 starts from zero)
    const int ssafe = (seg > 0) ? seg - 1 : 0;
    const float* sbp = ws + OFF_CS + ((size_t)ssafe * H + h) * 512;
    v8f S[4];
#pragma unroll
    for (int m = 0; m < 4; ++m) {
        S[m] = zero8();
#pragma unroll
        for (int r = 0; r < 8; ++r) {
            const int row = 16 * m + 8 * hw + r;
            S[m][r] = (seg > 0) ? sbp[row * 8 + b] : 0.0f;
        }
    }
#pragma unroll
    for (int m = 0; m < 4; ++m)
#pragma unroll
        for (int r = 0; r < 8; ++r)
            sbuf[cb][16 * m + 8 * hw + r] = S[m][r];
    __syncthreads();

    const float* xbase = x + (size_t)b * L * H + h;
    const int t00 = seg * LSEG;

    v2f xb[4];
    load_x(xbase, t00, hw, xb);                 // prime the pipeline

    for (int c = 0; c < CPS; ++c) {
        v2f xn[4];
        const int cn = (c + 1 < CPS) ? c + 1 : CPS - 1;
        load_x(xbase, t00 + cn * T, hw, xn);    // prefetch next chunk

        // Y = K @ Xc + R @ S_start
        v8f Y = zero8();
#pragma unroll
        for (int kk = 0; kk < 4; ++kk)
            Y = wmma4(ka[kk], xb[kk], Y);
#pragma unroll
        for (int kk = 0; kk < 16; ++kk) {
            v2f bs = mk2(sbuf[cb][4 * kk + 2 * hw], sbuf[cb][4 * kk + 2 * hw + 1]);
            Y = wmma4(ra[kk], bs, Y);
        }

        // exact GELU, store (all 32 lanes, 8 rows each)
        const int t0 = t00 + c * T;
#pragma unroll
        for (int r = 0; r < 8; ++r) {
            const int tau = 8 * hw + r;
            const float v = Y[r];
            const float g = 0.5f * v * (1.0f + erff(v * 0.70710678118654752f));
            out[((size_t)b * L + (t0 + tau)) * H + h] = g;
        }

        // S <- P (.) S + W @ Xc
#pragma unroll
        for (int m = 0; m < 4; ++m)
#pragma unroll
            for (int j = 0; j < 4; ++j) {
                const float re = S[m][2 * j], im = S[m][2 * j + 1];
                S[m][2 * j]     = pr[m][j] * re - pi[m][j] * im;
                S[m][2 * j + 1] = pr[m][j] * im + pi[m][j] * re;
            }
#pragma unroll
        for (int m = 0; m < 4; ++m)
#pragma unroll
            for (int kk = 0; kk < 4; ++kk)
                S[m] = wmma4(wa[m][kk], xb[kk], S[m]);

        // refresh LDS mirror for next chunk (per-wave LDS ops are in-order)
#pragma unroll
        for (int m = 0; m < 4; ++m)
#pragma unroll
            for (int r = 0; r < 8; ++r)
                sbuf[cb][16 * m + 8 * hw + r] = S[m][r];

#pragma unroll
        for (int kk = 0; kk < 4; ++kk) xb[kk] = xn[kk];
    }
}

// ---------------------------------------------------------------------------
extern "C" void kernel_launch(void* const* d_in, const int* in_sizes, int n_in,
                              void* d_out, int out_size, void* d_ws, size_t ws_size,
                              hipStream_t stream) {
    const float* x          = (const float*)d_in[0];
    const float* log_dt     = (const float*)d_in[1];
    const float* A_real_log = (const float*)d_in[2];
    const float* A_imag     = (const float*)d_in[3];
    const float* B_re       = (const float*)d_in[4];
    const float* B_im       = (const float*)d_in[5];
    const float* C_re       = (const float*)d_in[6];
    const float* C_im       = (const float*)d_in[7];
    const float* Dvec       = (const float*)d_in[8];
    float* out = (float*)d_out;
    float* ws  = (float*)d_ws;   // needs ~21.6 MB

    k_setup<<<dim3(H), dim3(32), 0, stream>>>(log_dt, A_real_log, A_imag,
                                              B_re, B_im, C_re, C_im, Dvec, ws);
    k_passA<<<dim3(H, NSP), dim3(32), 0, stream>>>(x, ws);
    k_combine<<<dim3(H), dim3(32), 0, stream>>>(log_dt, A_real_log, A_imag, ws, out);
    k_passB<<<dim3(H, NSP), dim3(32), 0, stream>>>(x, ws, out);
}